// RGCNWithLearnableAdj_43550968381997
// MI455X (gfx1250) — compile-verified
//
#include <hip/hip_runtime.h>
#include <hip/hip_bf16.h>

typedef __attribute__((ext_vector_type(2))) float v2f;
typedef __attribute__((ext_vector_type(8))) float v8f;

#define N_NODES 2048
#define IN_F 64
#define HID_F 32
#define OUT_F 16

// ---------------------------------------------------------------------------
// Kernel 1: column-sum of x [2048,64] -> colsum_x[64].  One workgroup of 1024
// threads = 32 waves; thread t sums column (t&63) over a 16-way row split,
// then a small LDS tree finishes the reduction. Fully coalesced 256B lines.
// ---------------------------------------------------------------------------
__global__ void k1_colsum_x(const float* __restrict__ x,
                            float* __restrict__ colsum_x) {
    __shared__ float part[16][IN_F];
    const int t   = threadIdx.x;        // 0..1023
    const int col = t & 63;
    const int g   = t >> 6;             // 0..15
    float acc = 0.f;
    for (int r = g; r < N_NODES; r += 16)
        acc += x[r * IN_F + col];
    part[g][col] = acc;
    __syncthreads();
    if (t < IN_F) {
        float s = 0.f;
        #pragma unroll
        for (int gg = 0; gg < 16; ++gg) s += part[gg][t];
        colsum_x[t] = s;
    }
}

// ---------------------------------------------------------------------------
// Kernel 2: h = relu(x @ loop_w1 + s1 + bias1), plus per-block partial
// colsum(h).  16 blocks x 256 threads (8 waves); each wave computes one
// 16-row tile with V_WMMA_F32_16X16X4_F32 (two N-tiles, K=64 in 16 steps).
// s1 = coeff1[0,0] * (colsum_x @ bases1[0]) is recomputed per block (2K MACs).
// ---------------------------------------------------------------------------
__global__ void k2_hidden(const float* __restrict__ x,        // [2048,64]
                          const float* __restrict__ bases1,   // [1,64,32]
                          const float* __restrict__ coeff1,   // [4,1]
                          const float* __restrict__ loop_w1,  // [64,32]
                          const float* __restrict__ bias1,    // [32]
                          const float* __restrict__ colsum_x, // [64]
                          float* __restrict__ h,              // [2048,32]
                          float* __restrict__ part_csh) {     // [16,32]
    __shared__ float sW[IN_F * HID_F];  // 8 KB
    __shared__ float s1[HID_F];
    __shared__ float scm[HID_F];

    const int t = threadIdx.x;          // 0..255
    for (int i = t; i < IN_F * HID_F; i += 256) sW[i] = loop_w1[i];
    if (t < HID_F) {
        float acc = 0.f;
        for (int k = 0; k < IN_F; ++k)
            acc += colsum_x[k] * bases1[k * HID_F + t];
        s1[t]  = coeff1[0] * acc + bias1[t];   // fold bias into broadcast term
        scm[t] = 0.f;
    }
    __syncthreads();

    const int wave = t >> 5;
    const int lane = t & 31;
    const int half = lane >> 4;          // K/row-half selector
    const int l16  = lane & 15;
    const int row0 = blockIdx.x * 128 + wave * 16;

    v8f acc0 = {};                       // cols 0..15
    v8f acc1 = {};                       // cols 16..31
    const float* xrow = x + (size_t)(row0 + l16) * IN_F;

    #pragma unroll
    for (int k0 = 0; k0 < IN_F; k0 += 4) {
        const int kb = k0 + 2 * half;
        v2f a;  a.x  = xrow[kb];                 a.y  = xrow[kb + 1];
        v2f b0; b0.x = sW[kb * HID_F + l16];     b0.y = sW[(kb + 1) * HID_F + l16];
        v2f b1; b1.x = sW[kb * HID_F + 16 + l16];b1.y = sW[(kb + 1) * HID_F + 16 + l16];
        acc0 = __builtin_amdgcn_wmma_f32_16x16x4_f32(false, a, false, b0,
                                                     (short)0, acc0, false, false);
        acc1 = __builtin_amdgcn_wmma_f32_16x16x4_f32(false, a, false, b1,
                                                     (short)0, acc1, false, false);
    }

    // Epilogue: bias/broadcast + ReLU, store h, accumulate partial colsum(h).
    float cs0 = 0.f, cs1 = 0.f;
    #pragma unroll
    for (int v = 0; v < 8; ++v) {
        const int r = row0 + 8 * half + v;
        float h0 = acc0[v] + s1[l16];       h0 = h0 > 0.f ? h0 : 0.f;
        float h1 = acc1[v] + s1[16 + l16];  h1 = h1 > 0.f ? h1 : 0.f;
        h[(size_t)r * HID_F + l16]      = h0;
        h[(size_t)r * HID_F + 16 + l16] = h1;
        cs0 += h0; cs1 += h1;
    }
    atomicAdd(&scm[l16], cs0);          // ds_add_f32, block-local
    atomicAdd(&scm[16 + l16], cs1);
    __syncthreads();
    if (t < HID_F) part_csh[blockIdx.x * HID_F + t] = scm[t];
}

// ---------------------------------------------------------------------------
// Kernel 3: out = h @ loop_w2 + s2 + bias2.  16 blocks x 256 threads; each
// block first reduces the 16 partial colsum(h) vectors and builds
// s2 = colsum(h) @ (sum_r coeff2[0,r]*bases2[r]) in LDS, then WMMAs K=32 in
// 8 steps into a single 16x16 N-tile per wave.
// ---------------------------------------------------------------------------
__global__ void k3_output(const float* __restrict__ h,        // [2048,32]
                          const float* __restrict__ bases2,   // [4,32,16]
                          const float* __restrict__ coeff2,   // [4,4]
                          const float* __restrict__ loop_w2,  // [32,16]
                          const float* __restrict__ bias2,    // [16]
                          const float* __restrict__ part_csh, // [16,32]
                          float* __restrict__ out) {          // [2048,16]
    __shared__ float sW[HID_F * OUT_F];  // 2 KB
    __shared__ float csh[HID_F];
    __shared__ float s2[OUT_F];

    const int t = threadIdx.x;
    for (int i = t; i < HID_F * OUT_F; i += 256) sW[i] = loop_w2[i];
    if (t < HID_F) {
        float acc = 0.f;
        #pragma unroll
        for (int b = 0; b < 16; ++b) acc += part_csh[b * HID_F + t];
        csh[t] = acc;
    }
    __syncthreads();
    if (t < OUT_F) {
        float acc = 0.f;
        for (int k = 0; k < HID_F; ++k) {
            float w = 0.f;
            #pragma unroll
            for (int r = 0; r < 4; ++r)
                w += coeff2[r] * bases2[(r * HID_F + k) * OUT_F + t];
            acc += csh[k] * w;
        }
        s2[t] = acc + bias2[t];
    }
    __syncthreads();

    const int wave = t >> 5;
    const int lane = t & 31;
    const int half = lane >> 4;
    const int l16  = lane & 15;
    const int row0 = blockIdx.x * 128 + wave * 16;

    v8f acc = {};
    const float* hrow = h + (size_t)(row0 + l16) * HID_F;
    #pragma unroll
    for (int k0 = 0; k0 < HID_F; k0 += 4) {
        const int kb = k0 + 2 * half;
        v2f a; a.x = hrow[kb];              a.y = hrow[kb + 1];
        v2f b; b.x = sW[kb * OUT_F + l16];  b.y = sW[(kb + 1) * OUT_F + l16];
        acc = __builtin_amdgcn_wmma_f32_16x16x4_f32(false, a, false, b,
                                                    (short)0, acc, false, false);
    }
    #pragma unroll
    for (int v = 0; v < 8; ++v) {
        const int r = row0 + 8 * half + v;
        out[(size_t)r * OUT_F + l16] = acc[v] + s2[l16];
    }
}

// ---------------------------------------------------------------------------
// Workspace layout (floats): h[2048*32] | colsum_x[64] | part_csh[16*32]
// ---------------------------------------------------------------------------
extern "C" void kernel_launch(void* const* d_in, const int* in_sizes, int n_in,
                              void* d_out, int out_size, void* d_ws, size_t ws_size,
                              hipStream_t stream) {
    const float* x       = (const float*)d_in[0];
    // d_in[1] = adj_matrix: provably unused (sigmoid(adj)>0 => complete graph)
    const float* bases1  = (const float*)d_in[2];
    const float* coeff1  = (const float*)d_in[3];
    const float* loop_w1 = (const float*)d_in[4];
    const float* bias1   = (const float*)d_in[5];
    const float* bases2  = (const float*)d_in[6];
    const float* coeff2  = (const float*)d_in[7];
    const float* loop_w2 = (const float*)d_in[8];
    const float* bias2   = (const float*)d_in[9];
    // d_in[10] = edge_type (ignored by the module)

    float* ws       = (float*)d_ws;
    float* h        = ws;                                  // 65536 floats
    float* colsum_x = ws + (size_t)N_NODES * HID_F;        // 64 floats
    float* part_csh = colsum_x + IN_F;                     // 512 floats

    k1_colsum_x<<<1, 1024, 0, stream>>>(x, colsum_x);
    k2_hidden<<<16, 256, 0, stream>>>(x, bases1, coeff1, loop_w1, bias1,
                                      colsum_x, h, part_csh);
    k3_output<<<16, 256, 0, stream>>>(h, bases2, coeff2, loop_w2, bias2,
                                      part_csh, (float*)d_out);
}